// GCN_30408368456212
// MI455X (gfx1250) — compile-verified
//
#include <hip/hip_runtime.h>
#include <math.h>

typedef __attribute__((ext_vector_type(2))) float v2f;
typedef __attribute__((ext_vector_type(8))) float v8f;

// ---------------------------------------------------------------------------
// Stage 0: FM sum/cross pooling.  ft0[n, 0:32]=sum_m x, ft0[n,32:64]=0.5(s^2-sum x^2)
// One wave (32 lanes == VL) per node; coalesced 128B row reads.
// ---------------------------------------------------------------------------
__global__ void __launch_bounds__(256) gcn_lv0(const float* __restrict__ feats,
                                               float* __restrict__ ft0, int n0) {
  int node = blockIdx.x * 8 + (threadIdx.x >> 5);
  int lane = threadIdx.x & 31;
  if (node >= n0) return;
  const float* p = feats + (size_t)node * 512 + lane;
  float s = 0.f, sq = 0.f;
#pragma unroll
  for (int m = 0; m < 16; ++m) {
    float x = p[m * 32];
    s += x;
    sq += x * x;
  }
  float cross = 0.5f * (s * s - sq);
  ft0[(size_t)node * 64 + lane] = s;
  ft0[(size_t)node * 64 + 32 + lane] = cross;
}

// ---------------------------------------------------------------------------
// Graph conv layer:  out = relu(concat(self+pool, self*pool) @ W + b)
//   ft_in : [*, 64] features (L2-resident gather table)
//   W     : [128, 64] row-major, b: [64]
// 8 waves per block; each wave owns one 16-row tile.
// GEMM done with V_WMMA_F32_16X16X4_F32 (fp32 exact), K=128 in 32 steps.
// LDS: W transposed with padded stride 132 (bank-conflict-free b64 reads),
//      per-wave A tile 16x128 (same padding).
// do_norm=1 adds the final L2 row-normalization via 16-lane shuffle reduce.
// ---------------------------------------------------------------------------
__global__ void __launch_bounds__(256) gcn_conv(
    const float* __restrict__ ft_in, const int* __restrict__ self_ids,
    const int* __restrict__ adj, const float* __restrict__ W,
    const float* __restrict__ bias, float* __restrict__ out, int n_nodes,
    int do_norm) {
  __shared__ __align__(16) float Wt[64 * 132];       // Wt[c*132 + k] = W[k][c]
  __shared__ __align__(16) float At[8 * 16 * 132];   // per-wave 16x128 (pad 132)

  // Stage W transposed into LDS (coalesced global reads).
  for (int i = threadIdx.x; i < 128 * 64; i += 256) {
    int k = i >> 6, c = i & 63;
    Wt[c * 132 + k] = W[i];
  }
  __syncthreads();

  const int wave = threadIdx.x >> 5;
  const int lane = threadIdx.x & 31;
  const int tile = blockIdx.x * 8 + wave;
  float* A = &At[wave * (16 * 132)];

  // Build A tile: row r = concat(self+pool, self*pool)  (128 cols)
#pragma unroll 1
  for (int r = 0; r < 16; ++r) {
    int node = tile * 16 + r;  // wave-uniform
    float a0 = 0.f, a1 = 0.f, m0 = 0.f, m1 = 0.f;
    if (node < n_nodes) {
      int sid = self_ids[node];
      const float* srow = ft_in + (size_t)sid * 64;
      float s0 = srow[lane], s1 = srow[32 + lane];
      float p0 = 0.f, p1 = 0.f;
      const int* arow = adj + (size_t)node * 32;
#pragma unroll 4
      for (int j = 0; j < 32; ++j) {
        const float* nrow = ft_in + (size_t)arow[j] * 64;
        p0 += nrow[lane];
        p1 += nrow[32 + lane];
      }
      a0 = s0 + p0;
      a1 = s1 + p1;
      m0 = s0 * p0;
      m1 = s1 * p1;
    }
    A[r * 132 + lane] = a0;
    A[r * 132 + 32 + lane] = a1;
    A[r * 132 + 64 + lane] = m0;
    A[r * 132 + 96 + lane] = m1;
  }
  __syncthreads();

  // WMMA fp32 GEMM: [16x128] @ [128x64] -> 4 accumulators of 16x16.
  // Fragment layout (16x16x4 f32): lane = {h=lane>>4, n=lane&15}
  //   A frag: row = n, K = 4*kt + 2*h + {0,1}
  //   B frag: col = n, K = 4*kt + 2*h + {0,1}  (from transposed Wt)
  const int n = lane & 15;
  const int h = lane >> 4;
  v8f acc[4] = {v8f{}, v8f{}, v8f{}, v8f{}};
#pragma unroll 1
  for (int kt = 0; kt < 32; ++kt) {
    v2f av = *(const v2f*)&A[n * 132 + kt * 4 + 2 * h];
#pragma unroll
    for (int nt = 0; nt < 4; ++nt) {
      v2f bv = *(const v2f*)&Wt[(nt * 16 + n) * 132 + kt * 4 + 2 * h];
      acc[nt] = __builtin_amdgcn_wmma_f32_16x16x4_f32(
          false, av, false, bv, (short)0, acc[nt], false, false);
    }
  }

  // Epilogue: bias + relu (lane holds rows v+8h, col nt*16+n).
  float z[4][8];
#pragma unroll
  for (int nt = 0; nt < 4; ++nt) {
    float bv = bias[nt * 16 + n];
#pragma unroll
    for (int v = 0; v < 8; ++v) z[nt][v] = fmaxf(acc[nt][v] + bv, 0.f);
  }

  if (do_norm) {
    // Per-row L2 norm: reduce over 16 lanes of each half-wave (cols), 4 nt each.
#pragma unroll
    for (int v = 0; v < 8; ++v) {
      float part = 0.f;
#pragma unroll
      for (int nt = 0; nt < 4; ++nt) part += z[nt][v] * z[nt][v];
      part += __shfl_xor(part, 1, 32);
      part += __shfl_xor(part, 2, 32);
      part += __shfl_xor(part, 4, 32);
      part += __shfl_xor(part, 8, 32);
      float inv = 1.f / fmaxf(sqrtf(part), 1e-12f);
#pragma unroll
      for (int nt = 0; nt < 4; ++nt) z[nt][v] *= inv;
    }
  }

#pragma unroll
  for (int v = 0; v < 8; ++v) {
    int node = tile * 16 + v + 8 * h;
    if (node < n_nodes) {
#pragma unroll
      for (int nt = 0; nt < 4; ++nt)
        out[(size_t)node * 64 + nt * 16 + n] = z[nt][v];
    }
  }
}

// ---------------------------------------------------------------------------
extern "C" void kernel_launch(void* const* d_in, const int* in_sizes, int n_in,
                              void* d_out, int out_size, void* d_ws,
                              size_t ws_size, hipStream_t stream) {
  (void)n_in;
  (void)out_size;
  (void)ws_size;
  const float* feats = (const float*)d_in[0];
  const int* l1ids = (const int*)d_in[1];
  const int* adj1 = (const int*)d_in[2];
  const int* pos2 = (const int*)d_in[3];
  const int* adj2 = (const int*)d_in[4];
  const float* W1 = (const float*)d_in[5];
  const float* b1 = (const float*)d_in[6];
  const float* W2 = (const float*)d_in[7];
  const float* b2 = (const float*)d_in[8];
  float* out = (float*)d_out;

  const int N0 = in_sizes[0] / (16 * 32);
  const int N1 = in_sizes[1];
  const int N2 = in_sizes[3];

  float* ft0 = (float*)d_ws;                 // [N0, 64]  (51.2 MB)
  float* ft1 = ft0 + (size_t)N0 * 64;        // [N1, 64]  (25.6 MB)

  gcn_lv0<<<(N0 + 7) / 8, 256, 0, stream>>>(feats, ft0, N0);

  int tiles1 = (N1 + 15) / 16;
  gcn_conv<<<(tiles1 + 7) / 8, 256, 0, stream>>>(ft0, l1ids, adj1, W1, b1, ft1,
                                                 N1, 0);

  int tiles2 = (N2 + 15) / 16;
  gcn_conv<<<(tiles2 + 7) / 8, 256, 0, stream>>>(ft1, pos2, adj2, W2, b2, out,
                                                 N2, 1);
}